// ModelNew_73315091744174
// MI455X (gfx1250) — compile-verified
//
#include <hip/hip_runtime.h>
#include <hip/hip_bf16.h>
#include <math.h>

// Problem: B=4096, IN=8192, OUT=8192, all fp32.
//   out[b] = log(exp(log(exp( (x[b,:]·colsum(W) + sum(bias)) / OUT ))))
// Algebraic collapse of the GEMM: sum_o(x·W[o,:]+b[o]) = x·colsum(W)+sum(b).
// Memory-bound: 256MB (W) + 128MB (x) ≈ 17us at 23.3 TB/s.
// Reductions go through V_WMMA_F32_16X16X4_F32 (exact fp32 math).
// Streaming loads use TH=NT so the one-pass 384MB stream doesn't thrash the
// 192MB L2; the reused 32KB w_sum stays resident for the GEMV.

#define B_     4096
#define IN_    8192
#define OUT_   8192
#define ROWSPLIT 8      // W row-range split for colsum partials
#define KSPLIT   8      // K split for GEMV partials

typedef __attribute__((ext_vector_type(2))) float v2f;
typedef __attribute__((ext_vector_type(8))) float v8f;

// ---------------------------------------------------------------------------
// Kernel 1: partial column-sum of W via WMMA.
//   A = ones(16x4), B = W tile (4 rows x 16 cols), D[m,n] += sum_k W[k,n].
// Grid: (IN/64, ROWSPLIT), block 128 (4 waves; each wave owns a 16-col strip).
// Dual accumulators break the WMMA RAW chain; merged at the end (linear).
// ---------------------------------------------------------------------------
__global__ void k_colsum_wmma(const float* __restrict__ W,
                              float* __restrict__ partial) {
    const int lane = threadIdx.x & 31;
    const int wave = threadIdx.x >> 5;
    const int h = lane >> 4;          // 0/1: which K-pair this half-wave holds
    const int n = lane & 15;          // column within strip
    const int colBase = blockIdx.x * 64 + wave * 16;
    const int rowsPer = OUT_ / ROWSPLIT;              // 1024 rows
    const int rowBase = blockIdx.y * rowsPer;

    v2f a; a.x = 1.0f; a.y = 1.0f;    // ones A-matrix
    v8f c0 = {}, c1 = {};

    const float* p = W + (size_t)(rowBase + h) * IN_ + colBase + n;
    #pragma unroll 2
    for (int it = 0; it < rowsPer / 8; ++it) {        // 8 rows / iter
        v2f b0, b1;
        b0.x = __builtin_nontemporal_load(p);                      // row +h
        b0.y = __builtin_nontemporal_load(p + (size_t)2 * IN_);    // row +2+h
        b1.x = __builtin_nontemporal_load(p + (size_t)4 * IN_);    // row +4+h
        b1.y = __builtin_nontemporal_load(p + (size_t)6 * IN_);    // row +6+h
        c0 = __builtin_amdgcn_wmma_f32_16x16x4_f32(false, a, false, b0,
                                                   (short)0, c0, false, false);
        c1 = __builtin_amdgcn_wmma_f32_16x16x4_f32(false, a, false, b1,
                                                   (short)0, c1, false, false);
        p += (size_t)8 * IN_;
    }
    v8f c = c0 + c1;
    // Every D row holds the same colsums; row M=0 lives in VGPR0, lanes 0-15.
    if (lane < 16)
        partial[(size_t)blockIdx.y * IN_ + colBase + lane] = c[0];
}

// ---------------------------------------------------------------------------
// Kernel 2: combine colsum partials (fixed order -> deterministic) and
// reduce bias (block 0 only, LDS tree).
// Grid: IN/256 = 32 blocks of 256.
// ---------------------------------------------------------------------------
__global__ void k_combine(const float* __restrict__ partial,
                          const float* __restrict__ bias,
                          float* __restrict__ w_sum,
                          float* __restrict__ b_sum) {
    const int i = blockIdx.x * 256 + threadIdx.x;
    float s = 0.f;
    #pragma unroll
    for (int r = 0; r < ROWSPLIT; ++r) s += partial[(size_t)r * IN_ + i];
    w_sum[i] = s;

    if (blockIdx.x == 0) {
        __shared__ float red[256];
        float bs = 0.f;
        for (int k = threadIdx.x; k < OUT_; k += 256) bs += bias[k];
        red[threadIdx.x] = bs;
        __syncthreads();
        for (int off = 128; off > 0; off >>= 1) {
            if (threadIdx.x < off) red[threadIdx.x] += red[threadIdx.x + off];
            __syncthreads();
        }
        if (threadIdx.x == 0) *b_sum = red[0];
    }
}

// ---------------------------------------------------------------------------
// Kernel 3: GEMV partials via WMMA.
//   A = x tile (16 rows x 4 K), B[k,n] = w_sum[k] broadcast across n.
//   D[m,n] += x[m,:]·w for every n; extract via C/D lane layout.
// A layout (32-bit A 16x4): lane = m + 16*h holds A[m,2h],A[m,2h+1]
//   -> one aligned 8-byte vector load per lane (nontemporal stream).
// w_sum loads stay RT: 32KB, heavily reused, lives in cache.
// Grid: (B/64, KSPLIT), block 128 (4 waves x 16 rows each) = 2048 waves.
// ---------------------------------------------------------------------------
__global__ void k_gemv_wmma(const float* __restrict__ x,
                            const float* __restrict__ w_sum,
                            float* __restrict__ part2) {
    const int lane = threadIdx.x & 31;
    const int wave = threadIdx.x >> 5;
    const int h = lane >> 4;
    const int m = lane & 15;
    const int rbase  = blockIdx.x * 64 + wave * 16;
    const int kchunk = IN_ / KSPLIT;                  // 1024
    const int kstart = blockIdx.y * kchunk;

    const float* px = x + (size_t)(rbase + m) * IN_ + kstart + 2 * h;
    const float* pw = w_sum + kstart + h;

    v8f c0 = {}, c1 = {};
    #pragma unroll 2
    for (int it = 0; it < kchunk / 8; ++it) {         // K=8 / iter
        v2f a0 = __builtin_nontemporal_load((const v2f*)px);
        v2f a1 = __builtin_nontemporal_load((const v2f*)(px + 4));
        v2f b0, b1;
        b0.x = pw[0]; b0.y = pw[2];                   // w[k+h], w[k+2+h]
        b1.x = pw[4]; b1.y = pw[6];
        c0 = __builtin_amdgcn_wmma_f32_16x16x4_f32(false, a0, false, b0,
                                                   (short)0, c0, false, false);
        c1 = __builtin_amdgcn_wmma_f32_16x16x4_f32(false, a1, false, b1,
                                                   (short)0, c1, false, false);
        px += 8; pw += 8;
    }
    v8f c = c0 + c1;
    // C/D layout: VGPR r: lanes0-15 -> M=r (N=lane), lanes16-31 -> M=8+r.
    float* dst = part2 + (size_t)blockIdx.y * B_ + rbase;
    if (lane == 0) {
        #pragma unroll
        for (int r = 0; r < 8; ++r) dst[r] = c[r];
    }
    if (lane == 16) {
        #pragma unroll
        for (int r = 0; r < 8; ++r) dst[8 + r] = c[r];
    }
}

// ---------------------------------------------------------------------------
// Kernel 4: finalize. Sum K-split partials (fixed order), add bias sum,
// divide by OUT, apply the reference's explicit log(exp(.)) chain twice.
// ---------------------------------------------------------------------------
__global__ void k_finalize(const float* __restrict__ part2,
                           const float* __restrict__ b_sum,
                           float* __restrict__ out) {
    const int b = blockIdx.x * 256 + threadIdx.x;
    float t = 0.f;
    #pragma unroll
    for (int ck = 0; ck < KSPLIT; ++ck) t += part2[(size_t)ck * B_ + b];
    float v = (t + *b_sum) * (1.0f / (float)OUT_);
    v = logf(expf(v));    // lse1 over size-1 dim
    v = logf(expf(v));    // lse2 over size-1 dim
    out[b] = v;
}

// ---------------------------------------------------------------------------
extern "C" void kernel_launch(void* const* d_in, const int* in_sizes, int n_in,
                              void* d_out, int out_size, void* d_ws, size_t ws_size,
                              hipStream_t stream) {
    const float* x    = (const float*)d_in[0];   // [4096, 8192]
    const float* W    = (const float*)d_in[1];   // [8192, 8192]
    const float* bias = (const float*)d_in[2];   // [8192]
    float* out = (float*)d_out;                  // [4096]

    // Workspace layout (all 16B-aligned):
    float* ws      = (float*)d_ws;
    float* part2   = ws;                               // KSPLIT * B      = 32768
    float* partial = part2 + (size_t)KSPLIT * B_;      // ROWSPLIT * IN   = 65536
    float* w_sum   = partial + (size_t)ROWSPLIT * IN_; // IN              = 8192
    float* b_sum   = w_sum + IN_;                      // 1

    k_colsum_wmma<<<dim3(IN_ / 64, ROWSPLIT), 128, 0, stream>>>(W, partial);
    k_combine    <<<dim3(IN_ / 256), 256, 0, stream>>>(partial, bias, w_sum, b_sum);
    k_gemv_wmma  <<<dim3(B_ / 64, KSPLIT), 128, 0, stream>>>(x, w_sum, part2);
    k_finalize   <<<dim3(B_ / 256), 256, 0, stream>>>(part2, b_sum, out);
}